// RBFKANHeadV2_69724499083647
// MI455X (gfx1250) — compile-verified
//
#include <hip/hip_runtime.h>
#include <hip/hip_bf16.h>
#include <cstdint>

typedef __attribute__((ext_vector_type(2))) float v2f;
typedef __attribute__((ext_vector_type(4))) float v4f;
typedef __attribute__((ext_vector_type(8))) float v8f;

#define IN_CH    256
#define HIDDEN   8
#define NB       8
#define IMG_HW   (160 * 160)        // 25600
#define NBATCH   16
#define PX_PER_WAVE   32            // two 16-px WMMA tiles per wave
#define WAVES_PER_BLK 8
#define PX_PER_BLK    (PX_PER_WAVE * WAVES_PER_BLK)   // 256; 25600 % 256 == 0

// D(16x16) = A(16x4) x B(4x16) + C, fp32 WMMA, K=256 in 64 steps.
// A = conv_w rows (hidden 0..7 real, rows 8..15 zero-padded in LDS -> no exec
// divergence in the loop). B = 16 contiguous pixels of x; two tiles share A.
// D: lane n (n<16) holds z[hidden 0..7] for pixel tile_base+n in acc[0..7].
__global__ __launch_bounds__(256) void rbfkan_head_kernel(
    const float* __restrict__ x,          // (B, IN_CH, H, W)
    const float* __restrict__ conv_w,     // (HIDDEN, IN_CH)
    const float* __restrict__ conv_b,     // (HIDDEN)
    const float* __restrict__ centers,    // 64
    const float* __restrict__ log_widths, // 64
    const float* __restrict__ out_weight, // (2, HIDDEN, NB) -> 128
    const float* __restrict__ out_bias,   // (2)
    float* __restrict__ out)              // (B, 2, H, W)
{
    __shared__ float w_lds[16 * IN_CH];   // 16 KB: rows 0..7 = conv_w, 8..15 = 0
    __shared__ v4f   post[HIDDEN * NB];   // {center, 1/width, w_out0, w_out1}

    const int tid = threadIdx.x;

    // Stage conv_w (2048 floats) + zero rows 8..15 (2048 floats), float4 each.
    {
        const v4f* src = (const v4f*)conv_w;
        v4f* dst = (v4f*)w_lds;
        dst[tid]       = src[tid];
        dst[tid + 256] = src[tid + 256];
        v4f zero = {0.f, 0.f, 0.f, 0.f};
        dst[tid + 512] = zero;
        dst[tid + 768] = zero;
    }
    // Precompute RBF constants: width = softplus(lw) + 0.001, store 1/width.
    if (tid < HIDDEN * NB) {
        float lw = log_widths[tid];
        float width = __logf(1.0f + __expf(lw)) + 0.001f;
        v4f p;
        p.x = centers[tid];
        p.y = 1.0f / width;
        p.z = out_weight[tid];         // out channel 0
        p.w = out_weight[64 + tid];    // out channel 1
        post[tid] = p;
    }
    __syncthreads();

    const int lane = tid & 31;
    const int wave = tid >> 5;
    const int px0  = blockIdx.x * PX_PER_BLK + wave * PX_PER_WAVE;
    const int b    = px0 / IMG_HW;            // tiles never straddle a batch image
    const int hw   = px0 % IMG_HW;

    const int n  = lane & 15;                 // B: pixel col / A: hidden row / D: col
    const int kb = (lane >> 4) << 1;          // K sub-pair: lanes 0-15 -> {0,1}, 16-31 -> {2,3}

    const float* xb  = x + (size_t)b * IN_CH * IMG_HW + hw;
    const float* xl0 = xb + n;                // tile 0: pixels px0 .. px0+15
    const float* xl1 = xb + 16 + n;           // tile 1: pixels px0+16 .. px0+31
    const float* wl  = &w_lds[n * IN_CH + kb];

    v8f acc0 = {0.f, 0.f, 0.f, 0.f, 0.f, 0.f, 0.f, 0.f};
    v8f acc1 = acc0;

#pragma unroll
    for (int s = 0; s < IN_CH / 4; ++s) {
        const int c = 4 * s + kb;

        // A operand: conv_w[row][c], conv_w[row][c+1] (zero-padded rows -> no branch)
        v2f amat = *(const v2f*)&wl[4 * s];

        // B operands: x[c][px], x[c+1][px] for each 16-px tile (channel stride IMG_HW)
        v2f b0, b1;
        b0.x = xl0[(size_t)c * IMG_HW];
        b0.y = xl0[(size_t)(c + 1) * IMG_HW];
        b1.x = xl1[(size_t)c * IMG_HW];
        b1.y = xl1[(size_t)(c + 1) * IMG_HW];

        if ((s & 7) == 0 && (s + 8) < IN_CH / 4) {
            __builtin_prefetch(&xl0[(size_t)(c + 32) * IMG_HW], 0, 1);
        }

        acc0 = __builtin_amdgcn_wmma_f32_16x16x4_f32(
            false, amat, false, b0, (short)0, acc0, false, false);
        acc1 = __builtin_amdgcn_wmma_f32_16x16x4_f32(
            false, amat, false, b1, (short)0, acc1, false, false);
    }

    // Redistribute: lanes 0-15 keep tile0 (pixel = lane), lanes 16-31 take tile1
    // (pixel = lane) via cross-lane shuffle. Then every lane owns one pixel.
    float z[HIDDEN];
#pragma unroll
    for (int r = 0; r < HIDDEN; ++r) {
        float t1 = __shfl(acc1[r], lane & 15);      // lanes 16-31 read lane-16's acc1
        z[r] = ((lane < 16) ? acc0[r] : t1) + conv_b[r];
    }

    // Per-lane RBF + 2-channel contraction, all 32 lanes active.
    float a0 = out_bias[0];
    float a1 = out_bias[1];
#pragma unroll
    for (int r = 0; r < HIDDEN; ++r) {
#pragma unroll
        for (int j = 0; j < NB; ++j) {
            v4f p = post[r * NB + j];
            float t  = (z[r] - p.x) * p.y;
            float ph = __expf(-t * t);
            a0 = fmaf(ph, p.z, a0);
            a1 = fmaf(ph, p.w, a1);
        }
    }

    // Full-wave contiguous 128B stores per output channel.
    float* ob = out + (size_t)b * 2 * IMG_HW + hw + lane;
    ob[0]      = a0;   // out channel 0
    ob[IMG_HW] = a1;   // out channel 1
}

extern "C" void kernel_launch(void* const* d_in, const int* in_sizes, int n_in,
                              void* d_out, int out_size, void* d_ws, size_t ws_size,
                              hipStream_t stream) {
    const float* x          = (const float*)d_in[0];
    const float* conv_w     = (const float*)d_in[1];
    const float* conv_b     = (const float*)d_in[2];
    const float* centers    = (const float*)d_in[3];
    const float* log_widths = (const float*)d_in[4];
    const float* out_weight = (const float*)d_in[5];
    const float* out_bias   = (const float*)d_in[6];
    float* out = (float*)d_out;

    dim3 grid((NBATCH * IMG_HW) / PX_PER_BLK);  // 409600 / 256 = 1600
    rbfkan_head_kernel<<<grid, 256, 0, stream>>>(
        x, conv_w, conv_b, centers, log_widths, out_weight, out_bias, out);
}